// TextTokenSelection_6150393168250
// MI455X (gfx1250) — compile-verified
//
#include <hip/hip_runtime.h>
#include <math.h>

#define B_ 128
#define N_ 512
#define D_ 512
#define C_ 256
#define K_ 4
#define S_ 500
#define SIGMA_ 0.05f
#define HALF_CNT 16384000u   // B*S*N/2

#define STR_K 66             // K-major LDS stride (even => b64-aligned pairs)
#define STR_S 260            // epilogue s-tile stride

typedef __attribute__((ext_vector_type(2))) float v2f;
typedef __attribute__((ext_vector_type(8))) float v8f;

__device__ __forceinline__ float gelu_exact(float v) {
    return 0.5f * v * (1.0f + erff(v * 0.70710678118654752f));
}

// ---------------------------------------------------------------- cls argmax
__global__ void cls_pos_kernel(const int* __restrict__ ids, int* __restrict__ cls) {
    const int b = blockIdx.x, ln = threadIdx.x;
    const int* row = ids + b * N_;
    int bv = -2147483647 - 1, bi = 0;
    for (int i = 0; i < 16; ++i) {           // ascending n => first-max kept
        int n = i * 32 + ln;
        int v = row[n];
        if (v > bv) { bv = v; bi = n; }
    }
    for (int off = 16; off; off >>= 1) {
        int ov = __shfl_xor(bv, off);
        int oi = __shfl_xor(bi, off);
        if (ov > bv || (ov == bv && oi < bi)) { bv = ov; bi = oi; }
    }
    if (ln == 0) cls[b] = bi;
}

// ------------------------------------------- GEMM1: gelu(LN(x)@w1 + b1) -> h
// block: 32 rows (flat B*N) x 256 cols; 8 waves x 4 accum tiles (no spill)
__global__ __launch_bounds__(256, 1) void gemm1_ln_gelu(
    const float* __restrict__ x, const float* __restrict__ lnw,
    const float* __restrict__ lnb, const float* __restrict__ w1,
    const float* __restrict__ b1, float* __restrict__ h) {
    __shared__ float meanL[32], rstdL[32];
    __shared__ float aL[32 * STR_K];          // [row][k] K-major
    __shared__ float bT[256 * STR_K];         // [col][k] K-major (transposed)
    const int t = threadIdx.x, ln = t & 31, wv = t >> 5;
    const int row0 = blockIdx.x * 32;

    // pass 1: per-row mean / rstd (wave per 4 rows)
    for (int rr = 0; rr < 4; ++rr) {
        const int row = wv * 4 + rr;
        const float* xr = x + (size_t)(row0 + row) * D_;
        float s = 0.f, q = 0.f;
        for (int m = 0; m < 16; ++m) { float v = xr[m * 32 + ln]; s += v; q += v * v; }
        for (int off = 16; off; off >>= 1) { s += __shfl_xor(s, off); q += __shfl_xor(q, off); }
        if (ln == 0) {
            float mu = s * (1.f / 512.f);
            meanL[row] = mu;
            rstdL[row] = rsqrtf(q * (1.f / 512.f) - mu * mu + 1e-5f);
        }
    }
    __syncthreads();

    const int mt = wv & 1, cq = wv >> 1;      // row tile (2), col quarter (4)
    v8f acc[4];
    for (int j = 0; j < 4; ++j)
        for (int r = 0; r < 8; ++r) acc[j][r] = 0.f;

    for (int kc = 0; kc < 8; ++kc) {
        __syncthreads();
        for (int e = t; e < 32 * 64; e += 256) {
            int r = e >> 6, kk = e & 63, kg = kc * 64 + kk;
            float v = x[(size_t)(row0 + r) * D_ + kg];
            aL[r * STR_K + kk] = (v - meanL[r]) * rstdL[r] * lnw[kg] + lnb[kg];
        }
        for (int e = t; e < 64 * 256; e += 256) {
            int kk = e >> 8, c = e & 255;     // coalesced w1 read
            bT[c * STR_K + kk] = w1[(size_t)(kc * 64 + kk) * C_ + c];
        }
        __syncthreads();
        const int mrow = mt * 16 + (ln & 15);
        const int koff = (ln < 16) ? 0 : 2;   // ISA: VGPR0 = K0/K2, VGPR1 = K1/K3
        for (int k4 = 0; k4 < 16; ++k4) {
            const int kb = k4 * 4 + koff;
            v2f a = *(const v2f*)(aL + mrow * STR_K + kb);     // one ds_load_b64
#pragma unroll
            for (int j = 0; j < 4; ++j) {
                const int cb = (cq * 4 + j) * 16 + (ln & 15);
                v2f bb = *(const v2f*)(bT + cb * STR_K + kb);  // one ds_load_b64
                acc[j] = __builtin_amdgcn_wmma_f32_16x16x4_f32(
                    false, a, false, bb, (short)0, acc[j], false, false);
            }
        }
    }
    // epilogue: +bias, gelu, store (D layout: VGPR r = rows r / r+8)
    const int rbase = row0 + mt * 16 + ((ln < 16) ? 0 : 8);
    const int cl = ln & 15;
    for (int j = 0; j < 4; ++j) {
        const int gcol = (cq * 4 + j) * 16 + cl;
        const float bias = b1[gcol];
#pragma unroll
        for (int r = 0; r < 8; ++r)
            h[(size_t)(rbase + r) * C_ + gcol] = gelu_exact(acc[j][r] + bias);
    }
}

// -------------------------------- gc[b,:] = h[b,cls,:] @ w2[256:512,:]
__global__ void gcontrib_kernel(const float* __restrict__ h, const int* __restrict__ cls,
                                const float* __restrict__ w2, float* __restrict__ gcout) {
    __shared__ float gL[256];
    const int b = blockIdx.x, t = threadIdx.x;
    const int cp = cls[b];
    gL[t] = h[((size_t)b * N_ + cp) * C_ + t];
    __syncthreads();
    float a = 0.f;
    for (int j = 0; j < 256; ++j) a += gL[j] * w2[(size_t)(256 + j) * C_ + t];
    gcout[b * C_ + t] = a;
}

// ----- GEMM2 fused: s = gelu(h@w2[:256] + gc); score = sigmoid(s@w3)*mask'
// block: 32 rows x 256 cols; reuses bT as the epilogue s-tile
__global__ __launch_bounds__(256, 1) void gemm2_score(
    const float* __restrict__ h, const float* __restrict__ w2,
    const float* __restrict__ w3, const float* __restrict__ gc,
    const float* __restrict__ amask, float* __restrict__ wscore) {
    __shared__ float aL[32 * STR_K];
    __shared__ float bT[256 * STR_K];         // >= 32*STR_S, reused as s-tile
    __shared__ float w3L[256];
    __shared__ float gcL[256];
    __shared__ float red[32 * 8];
    const int t = threadIdx.x, ln = t & 31, wv = t >> 5;
    const int row0 = blockIdx.x * 32;
    const int b = row0 >> 9;                  // 32-row block inside one batch
    w3L[t] = w3[t];
    gcL[t] = gc[b * C_ + t];

    const int mt = wv & 1, cq = wv >> 1;
    v8f acc[4];
    for (int j = 0; j < 4; ++j)
        for (int r = 0; r < 8; ++r) acc[j][r] = 0.f;

    for (int kc = 0; kc < 4; ++kc) {
        __syncthreads();
        for (int e = t; e < 32 * 64; e += 256) {
            int r = e >> 6, kk = e & 63;
            aL[r * STR_K + kk] = h[(size_t)(row0 + r) * C_ + kc * 64 + kk];
        }
        for (int e = t; e < 64 * 256; e += 256) {
            int kk = e >> 8, c = e & 255;
            bT[c * STR_K + kk] = w2[(size_t)(kc * 64 + kk) * C_ + c];
        }
        __syncthreads();
        const int mrow = mt * 16 + (ln & 15);
        const int koff = (ln < 16) ? 0 : 2;
        for (int k4 = 0; k4 < 16; ++k4) {
            const int kb = k4 * 4 + koff;
            v2f a = *(const v2f*)(aL + mrow * STR_K + kb);
#pragma unroll
            for (int j = 0; j < 4; ++j) {
                const int cb = (cq * 4 + j) * 16 + (ln & 15);
                v2f bb = *(const v2f*)(bT + cb * STR_K + kb);
                acc[j] = __builtin_amdgcn_wmma_f32_16x16x4_f32(
                    false, a, false, bb, (short)0, acc[j], false, false);
            }
        }
    }
    __syncthreads();
    {   // write gelu(acc + gc) into reused bT as s-tile [32][STR_S]
        const int rb = mt * 16 + ((ln < 16) ? 0 : 8);
        const int cl = ln & 15;
        for (int j = 0; j < 4; ++j) {
            const int gcol = (cq * 4 + j) * 16 + cl;
#pragma unroll
            for (int r = 0; r < 8; ++r)
                bT[(rb + r) * STR_S + gcol] = gelu_exact(acc[j][r] + gcL[gcol]);
        }
    }
    __syncthreads();
    {   // 8 threads per row reduce against w3
        const int row = t >> 3, part = t & 7;
        const float* srow = &bT[row * STR_S];
        float p = 0.f;
        for (int i = part * 32; i < part * 32 + 32; ++i) p += srow[i] * w3L[i];
        red[row * 8 + part] = p;
    }
    __syncthreads();
    if (t < 32) {
        float z = 0.f;
        for (int i = 0; i < 8; ++i) z += red[t * 8 + i];
        float sig = 1.f / (1.f + expf(-z));
        const int n = (row0 + t) & 511;
        float m = (n < 511) ? amask[b * N_ + n + 1] : 0.f;
        wscore[b * N_ + n] = sig * m;
    }
}

// ------------------------------------------------- JAX threefry2x32, key(12345)
__device__ __forceinline__ void threefry2x32(uint32_t x0, uint32_t x1,
                                             uint32_t& o0, uint32_t& o1) {
    const uint32_t k0 = 0u, k1 = 12345u;
    const uint32_t k2 = 0x1BD11BDAu ^ k0 ^ k1;
    x0 += k0; x1 += k1;
#define TF_R(r) { x0 += x1; x1 = (x1 << (r)) | (x1 >> (32 - (r))); x1 ^= x0; }
    TF_R(13) TF_R(15) TF_R(26) TF_R(6)
    x0 += k1; x1 += k2 + 1u;
    TF_R(17) TF_R(29) TF_R(16) TF_R(24)
    x0 += k2; x1 += k0 + 2u;
    TF_R(13) TF_R(15) TF_R(26) TF_R(6)
    x0 += k0; x1 += k1 + 3u;
    TF_R(17) TF_R(29) TF_R(16) TF_R(24)
    x0 += k1; x1 += k2 + 4u;
    TF_R(13) TF_R(15) TF_R(26) TF_R(6)
    x0 += k2; x1 += k0 + 5u;
#undef TF_R
    o0 = x0; o1 = x1;
}

__device__ __forceinline__ float bits_to_normal(uint32_t bits) {
    // JAX uniform(lo=nextafter(-1,0), hi=1) then sqrt(2)*erfinv
    float f = __uint_as_float((bits >> 9) | 0x3f800000u) - 1.0f;   // [0,1)
    const float lo = -0.99999994f;
    float u = fmaxf(lo, f * (1.0f - lo) + lo);
    return 1.41421356237f * erfinvf(u);
}

// iterative top-4 over 512 vals (value desc, index asc tie-break = lax.top_k)
__device__ __forceinline__ void top4_accum(float (&v)[16], int ln, float* indRow) {
    int sel[4];
#pragma unroll
    for (int k = 0; k < 4; ++k) {
        float bv = -__builtin_inff(); int bi = 0x7fffffff;
#pragma unroll
        for (int i = 0; i < 16; ++i) {   // ascending n keeps smallest-index tie
            if (v[i] > bv) { bv = v[i]; bi = i * 32 + ln; }
        }
        for (int off = 16; off; off >>= 1) {
            float ov = __shfl_xor(bv, off);
            int   oi = __shfl_xor(bi, off);
            if (ov > bv || (ov == bv && oi < bi)) { bv = ov; bi = oi; }
        }
        sel[k] = bi;
        if ((bi & 31) == ln) {
            const int slot = bi >> 5;
#pragma unroll
            for (int i = 0; i < 16; ++i) if (i == slot) v[i] = -__builtin_inff();
        }
    }
    // sort 4 indices ascending (jnp.sort before one_hot)
    auto cswap = [](int& a, int& b) { if (a > b) { int tt = a; a = b; b = tt; } };
    cswap(sel[0], sel[1]); cswap(sel[2], sel[3]);
    cswap(sel[0], sel[2]); cswap(sel[1], sel[3]); cswap(sel[1], sel[2]);
    if (ln == 0) {
#pragma unroll
        for (int k = 0; k < 4; ++k) atomicAdd(&indRow[k * N_ + sel[k]], 1.0f);
    }
}

// one wave = one sample s for batch pair (b, b+64); threefry pair reuse
__global__ __launch_bounds__(256, 1) void topk_kernel(const float* __restrict__ wscore,
                                                      float* __restrict__ ind) {
    const int gw = blockIdx.x * 8 + (threadIdx.x >> 5);  // 0..31999
    const int ln = threadIdx.x & 31;
    const int b = gw / S_, s = gw % S_;                  // b in 0..63
    const float* s1 = wscore + b * N_;
    const float* s2 = wscore + (b + 64) * N_;
    const uint32_t base = ((uint32_t)(b * S_ + s)) << 9;
    float vA[16], vB[16];
#pragma unroll
    for (int i = 0; i < 16; ++i) {
        const uint32_t n = (uint32_t)(i * 32 + ln);
        uint32_t o0, o1;
        threefry2x32(base + n, base + n + HALF_CNT, o0, o1);
        vA[i] = s1[n] + SIGMA_ * bits_to_normal(o0);
        vB[i] = s2[n] + SIGMA_ * bits_to_normal(o1);
    }
    top4_accum(vA, ln, ind + (size_t)b * K_ * N_);
    top4_accum(vB, ln, ind + (size_t)(b + 64) * K_ * N_);
}

// -------------------- out[b] = [x[b,cls], (ind/500) @ x[b]]  (B,5,D)
__global__ __launch_bounds__(256, 1) void select_kernel(
    const float* __restrict__ x, const float* __restrict__ ind,
    const int* __restrict__ cls, float* __restrict__ out) {
    __shared__ float iL[K_ * N_];
    const int b = blockIdx.x, t = threadIdx.x;
    for (int e = t; e < K_ * N_; e += 256) iL[e] = ind[(size_t)b * K_ * N_ + e];
    __syncthreads();
    const int d0 = t * 2;
    float a0[4] = {0.f, 0.f, 0.f, 0.f}, a1[4] = {0.f, 0.f, 0.f, 0.f};
    const float* xb = x + (size_t)b * N_ * D_;
    for (int n = 0; n < N_; ++n) {
        float2 xv = *(const float2*)(xb + (size_t)n * D_ + d0);
#pragma unroll
        for (int k = 0; k < 4; ++k) {
            float w = iL[k * N_ + n];
            a0[k] += w * xv.x; a1[k] += w * xv.y;
        }
    }
    float2 cx = *(const float2*)(xb + (size_t)cls[b] * D_ + d0);
    float* ob = out + (size_t)b * 5 * D_;
    *(float2*)(ob + d0) = cx;
    const float inv = 1.0f / (float)S_;
#pragma unroll
    for (int k = 0; k < 4; ++k) {
        float2 r; r.x = a0[k] * inv; r.y = a1[k] * inv;
        *(float2*)(ob + (size_t)(1 + k) * D_ + d0) = r;
    }
}

extern "C" void kernel_launch(void* const* d_in, const int* in_sizes, int n_in,
                              void* d_out, int out_size, void* d_ws, size_t ws_size,
                              hipStream_t stream) {
    (void)in_sizes; (void)n_in; (void)out_size; (void)ws_size;
    const float* x     = (const float*)d_in[0];
    const int*   ids   = (const int*)  d_in[1];
    const float* amask = (const float*)d_in[2];
    const float* lnw   = (const float*)d_in[3];
    const float* lnb   = (const float*)d_in[4];
    const float* w1    = (const float*)d_in[5];
    const float* b1    = (const float*)d_in[6];
    const float* w2    = (const float*)d_in[7];
    const float* w3    = (const float*)d_in[8];
    float* out = (float*)d_out;

    char* wsb = (char*)d_ws;
    int*   cls    = (int*)wsb;                                        // 128 ints
    float* gc     = (float*)(wsb + 1024);                             // 128*256
    float* wscore = (float*)(wsb + 1024 + 131072);                    // 128*512
    float* ind    = (float*)(wsb + 1024 + 131072 + 262144);           // 128*4*512
    float* h      = (float*)(wsb + 1024 + 131072 + 262144 + 1048576); // 65536*256

    hipMemsetAsync(ind, 0, sizeof(float) * B_ * K_ * N_, stream);
    cls_pos_kernel<<<B_, 32, 0, stream>>>(ids, cls);
    gemm1_ln_gelu<<<(B_ * N_) / 32, 256, 0, stream>>>(x, lnw, lnb, w1, b1, h);
    gcontrib_kernel<<<B_, 256, 0, stream>>>(h, cls, w2, gc);
    gemm2_score<<<(B_ * N_) / 32, 256, 0, stream>>>(h, w2, w3, gc, amask, wscore);
    topk_kernel<<<(64 * S_) / 8, 256, 0, stream>>>(wscore, ind);
    select_kernel<<<B_, 256, 0, stream>>>(x, ind, cls, out);
}